// RelAttLayer_12146167513336
// MI455X (gfx1250) — compile-verified
//
#include <hip/hip_runtime.h>

// CDNA5 / gfx1250, wave32. f32 WMMA path: V_WMMA_F32_16X16X4_F32 chained x4 (K=16).
typedef __attribute__((ext_vector_type(2))) float v2f;
typedef __attribute__((ext_vector_type(8))) float v8f;

__global__ void k_zero_f32(float* __restrict__ p, int n) {
  int i = blockIdx.x * blockDim.x + threadIdx.x;
  if (i < n) p[i] = 0.0f;
}

__global__ void k_zero_i32(int* __restrict__ p, int n) {
  int i = blockIdx.x * blockDim.x + threadIdx.x;
  if (i < n) p[i] = 0;
}

// W_rel (scrambled-reshape basis combine) + folded attention vectors a_src/a_e/a_dst.
// wrel flat index k = x*(R*16) + r*16 + z  == res layout; consumers read it as [R,16,16].
__global__ void k_prep(const float* __restrict__ weight, const float* __restrict__ w_comp,
                       const float* __restrict__ Ws, const float* __restrict__ Wa,
                       float* __restrict__ wrel, float* __restrict__ avec, int R, int B) {
  int k = blockIdx.x * blockDim.x + threadIdx.x;
  int total = 16 * R * 16;
  if (k < total) {
    int x = k / (R * 16);
    int r = (k / 16) % R;
    int z = k & 15;
    float acc = 0.0f;
    for (int b = 0; b < B; ++b)
      acc += w_comp[r * B + b] * weight[x * B * 16 + b * 16 + z];
    wrel[k] = acc;
  }
  if (k < 48) {  // a[part][i] = sum_o Wa[0, part*16+o] * Ws[o, i]
    int part = k >> 4, i = k & 15;
    float acc = 0.0f;
    for (int o = 0; o < 16; ++o)
      acc += Wa[part * 16 + o] * Ws[o * 16 + i];
    avec[k] = acc;
  }
}

__global__ void k_hist(const int* __restrict__ rel, int* __restrict__ hist, int E) {
  int e = blockIdx.x * blockDim.x + threadIdx.x;
  if (e < E) atomicAdd(&hist[rel[e]], 1);
}

__global__ void k_scan(const int* __restrict__ hist, int* __restrict__ cursor, int R) {
  if (blockIdx.x == 0 && threadIdx.x == 0) {
    int run = 0;
    for (int r = 0; r < R; ++r) { cursor[r] = run; run += hist[r]; }
  }
}

__global__ void k_scatter(const int* __restrict__ rel, int* __restrict__ cursor,
                          int* __restrict__ sorted, int E) {
  int e = blockIdx.x * blockDim.x + threadIdx.x;
  if (e < E) {
    int p = atomicAdd(&cursor[rel[e]], 1);
    sorted[p] = e;
  }
}

// One wave per 16-edge tile of the relation-sorted edge list.
__global__ void k_main(const float* __restrict__ h, const float* __restrict__ he,
                       const int* __restrict__ src, const int* __restrict__ dst,
                       const int* __restrict__ rel, const int* __restrict__ sorted,
                       const float* __restrict__ wrel, const float* __restrict__ avec,
                       float* __restrict__ out, int E, int ntiles) {
  int wavesPerBlock = blockDim.x >> 5;
  int tile = blockIdx.x * wavesPerBlock + (threadIdx.x >> 5);
  if (tile >= ntiles) return;              // wave-uniform
  int lane = threadIdx.x & 31;
  int lo = lane & 15, hi = lane >> 4;
  int e0 = tile * 16;

  int s = 0, d = 0, r = 0;
  float esc = 0.0f;
  bool own = (hi == 0) && (e0 + lo < E);   // lanes 0-15 each own one edge
  if (own) {
    int eid = sorted[e0 + lo];
    s = src[eid]; d = dst[eid]; r = rel[eid];
    const float* hs = h + (size_t)s * 16;
    const float* hd = h + (size_t)d * 16;
    const float* hx = he + (size_t)eid * 16;
    float acc = 0.0f;
#pragma unroll
    for (int i = 0; i < 16; ++i)
      acc += avec[i] * hs[i] + avec[16 + i] * hx[i] + avec[32 + i] * hd[i];
    esc = acc;                             // attention score (folded Wa.Ws form)
  }

  int r0 = __shfl(r, 0, 32);
  bool full = (e0 + 16 <= E);
  int ok = (!own) || (r == r0);
  if (full && __all(ok)) {
    // ---- WMMA fast path: msg(16x16) = (esc .* H_src)(16x16) x W_rel[r0](16x16), f32 ----
    // A 16x4 f32 layout (ISA 7.12.2): lanes0-15 rows M=0..15; VGPR0 = K0|K2, VGPR1 = K1|K3.
    // B 4x16 f32: row-striped across lanes: VGPR0 = row0|row2, VGPR1 = row1|row3.
    // Attention score folded into A rows: (h.W)*e == (e*h).W, saves post-WMMA
    // bpermute+mul on the critical path into the atomics.
    int   sm   = __shfl(s, lo, 32);                     // src of edge M=lo
    float escm = __shfl(esc, lo, 32);                   // score of edge M=lo
    const float* arow  = h + (size_t)sm * 16 + 2 * hi;  // this lane's K-pair base
    const float* bbase = wrel + (size_t)r0 * 256 + (2 * hi) * 16 + lo;
    v8f c = {0.f, 0.f, 0.f, 0.f, 0.f, 0.f, 0.f, 0.f};
#pragma unroll
    for (int k = 0; k < 4; ++k) {                       // K slices 4k..4k+3
      v2f a; a.x = arow[4 * k] * escm; a.y = arow[4 * k + 1] * escm;
      v2f b; b.x = bbase[k * 64];      b.y = bbase[k * 64 + 16];
      c = __builtin_amdgcn_wmma_f32_16x16x4_f32(false, a, false, b, (short)0, c,
                                                false, false);
    }
    // D layout: VGPR q -> M = q + 8*hi, N = lo. Scatter-add (score already applied).
#pragma unroll
    for (int q = 0; q < 8; ++q) {
      int m = q + 8 * hi;
      int dm = __shfl(d, m, 32);
      atomicAdd(&out[(size_t)dm * 16 + lo], c[q]);
    }
  } else {
    // ---- scalar fallback: relation-boundary / partial tiles only (<=89 of 50000) ----
    if (own) {
      float hv[16];
      const float* hs = h + (size_t)s * 16;
#pragma unroll
      for (int i = 0; i < 16; ++i) hv[i] = hs[i];
      const float* W = wrel + (size_t)r * 256;
#pragma unroll 4
      for (int o = 0; o < 16; ++o) {
        float acc = 0.0f;
#pragma unroll
        for (int i = 0; i < 16; ++i) acc += hv[i] * W[i * 16 + o];
        atomicAdd(&out[(size_t)d * 16 + o], acc * esc);
      }
    }
  }
}

extern "C" void kernel_launch(void* const* d_in, const int* in_sizes, int n_in,
                              void* d_out, int out_size, void* d_ws, size_t ws_size,
                              hipStream_t stream) {
  const float* h      = (const float*)d_in[0];
  const float* he     = (const float*)d_in[1];
  const float* Ws     = (const float*)d_in[2];
  const float* Wa     = (const float*)d_in[3];
  const float* weight = (const float*)d_in[4];
  const float* w_comp = (const float*)d_in[5];
  const int*   src    = (const int*)d_in[6];
  const int*   dst    = (const int*)d_in[7];
  const int*   rel    = (const int*)d_in[8];
  float* out = (float*)d_out;

  int E = in_sizes[1] / 16;        // he is [E,16]
  int B = in_sizes[4] / 256;       // weight is [B,16,16]
  int R = in_sizes[5] / B;         // w_comp is [R,B]

  // workspace layout
  float* wrel = (float*)d_ws;                  // R*256 f32
  float* avec = wrel + (size_t)R * 256;        // 48 f32
  int* hist   = (int*)(avec + 48);             // R
  int* cursor = hist + R;                      // R
  int* sorted = cursor + R;                    // E
  (void)ws_size; (void)n_in;

  k_zero_f32<<<(out_size + 255) / 256, 256, 0, stream>>>(out, out_size);
  k_zero_i32<<<(R + 255) / 256, 256, 0, stream>>>(hist, R);

  int prepN = 16 * R * 16;
  k_prep<<<(prepN + 255) / 256, 256, 0, stream>>>(weight, w_comp, Ws, Wa, wrel, avec, R, B);

  k_hist<<<(E + 255) / 256, 256, 0, stream>>>(rel, hist, E);
  k_scan<<<1, 1, 0, stream>>>(hist, cursor, R);
  k_scatter<<<(E + 255) / 256, 256, 0, stream>>>(rel, cursor, sorted, E);

  int ntiles = (E + 15) / 16;
  int wavesPerBlock = 8;                       // 256 threads = 8 wave32
  int nblocks = (ntiles + wavesPerBlock - 1) / wavesPerBlock;
  k_main<<<nblocks, wavesPerBlock * 32, 0, stream>>>(h, he, src, dst, rel, sorted,
                                                     wrel, avec, out, E, ntiles);
}